// GDC_11776800326009
// MI455X (gfx1250) — compile-verified
//
#include <hip/hip_runtime.h>
#include <hip/hip_bf16.h>
#include <math.h>

// Problem constants (match reference)
#define NN   100000
#define EE   1600000
#define FIN  256
#define HID  64
#define NCLS 40
#define NC_PAD 48        // 3 x 16 column tiles for GEMM2
#define TPB1 5           // row tiles per block in GEMM1 (6250 tiles = 1250 blocks x 5)

typedef float v2f __attribute__((ext_vector_type(2)));
typedef float v8f __attribute__((ext_vector_type(8)));

// LDS byte offset: generic LDS address keeps the aperture in the high 32 bits,
// the low 32 bits are the LDS offset (ISA 10.2 aperture rules).
__device__ __forceinline__ unsigned lds_off(const void* p) {
  return (unsigned)(uintptr_t)p;
}

// Async 16B copy global -> LDS (ASYNCcnt-tracked, no VGPR data path).
__device__ __forceinline__ void async_copy_b128(unsigned lds_addr, const void* gaddr) {
  unsigned long long ga = (unsigned long long)(uintptr_t)gaddr;
  asm volatile("global_load_async_to_lds_b128 %0, %1, off"
               :: "v"(lds_addr), "v"(ga) : "memory");
}

__device__ __forceinline__ void wait_async0() {
  asm volatile("s_wait_asynccnt 0" ::: "memory");
}

// ---------------------------------------------------------------- zero fill
__global__ void gcn_zero(float* __restrict__ p, long long count) {
  long long i = (long long)blockIdx.x * blockDim.x + threadIdx.x;
  long long stride = (long long)gridDim.x * blockDim.x;
  for (; i < count; i += stride) p[i] = 0.0f;
}

// ---------------------------------------------------------------- degree
__global__ void gcn_deg(const long long* __restrict__ ei,
                        const float* __restrict__ ea,
                        float* __restrict__ deg) {
  int i = blockIdx.x * blockDim.x + threadIdx.x;
  if (i >= EE + NN) return;
  int d; float w;
  if (i < EE) { d = (int)ei[EE + i]; w = ea[i]; }
  else        { d = i - EE;          w = 1.0f;  }
  atomicAdd(&deg[d], w);
}

// deg -> D^{-1/2} in place
__global__ void gcn_dinv(float* __restrict__ deg) {
  int i = blockIdx.x * blockDim.x + threadIdx.x;
  if (i >= NN) return;
  float d = deg[i];
  deg[i] = (d > 0.0f) ? rsqrtf(fmaxf(d, 1e-30f)) : 0.0f;
}

// ---------------------------------------------------------------- GEMM1: h1 = x @ W1  (fp32 WMMA 16x16x4)
// 128 threads = 4 waves; wave w owns one 16-col tile of the 64 cols.
// Each block processes TPB1 row tiles: W1 staged in LDS once (async),
// x tiles double-buffered with GLOBAL_LOAD_ASYNC_TO_LDS_B128 so the
// next tile's fetch overlaps the 64-WMMA K-loop of the current tile.
__global__ __launch_bounds__(128) void gcn_gemm1(const float* __restrict__ x,
                                                 const float* __restrict__ W1,
                                                 float* __restrict__ h1) {
  __shared__ float lw[FIN * HID];      // 64 KB: W1
  __shared__ float lx[2][16 * FIN];    // 2 x 16 KB: x tile double buffer
  const int tid  = threadIdx.x;
  const int wave = tid >> 5;           // 0..3 -> column tile
  const int lane = tid & 31;
  const int tile0 = blockIdx.x * TPB1;

  // async-stage W1 (4096 float4 / 128 threads = 32 per thread, uniform)
  for (int i = tid; i < (FIN * HID) / 4; i += 128)
    async_copy_b128(lds_off(&lw[i * 4]), W1 + i * 4);
  // async-stage first x tile (1024 float4 / 128 = 8 per thread)
  {
    const float* src = x + (size_t)tile0 * 16 * FIN;
    for (int i = tid; i < (16 * FIN) / 4; i += 128)
      async_copy_b128(lds_off(&lx[0][i * 4]), src + i * 4);
  }
  wait_async0();
  __syncthreads();

  // WMMA f32 16x16x4 fragment mapping (ISA 7.12.2, 32-bit A 16x4):
  //   lanes 0-15:  VGPR0=K0, VGPR1=K1 (M=lane);  lanes 16-31: VGPR0=K2, VGPR1=K3
  const int half = lane >> 4;          // 0 | 1
  const int r    = lane & 15;
  const int col0 = wave * 16;

  int cur = 0;
  for (int t = 0; t < TPB1; ++t) {
    const int row0 = (tile0 + t) * 16;

    // prefetch next tile into the other buffer (overlaps with compute)
    if (t + 1 < TPB1) {
      const float* src = x + (size_t)(row0 + 16) * FIN;
      for (int i = tid; i < (16 * FIN) / 4; i += 128)
        async_copy_b128(lds_off(&lx[cur ^ 1][i * 4]), src + i * 4);
    }

    const float* xb = lx[cur];
    v8f c = {};
    for (int kb = 0; kb < FIN; kb += 4) {
      const int ka = kb + half * 2;
      v2f a, b;
      a.x = xb[r * FIN + ka];
      a.y = xb[r * FIN + ka + 1];
      b.x = lw[ka * HID + col0 + r];
      b.y = lw[(ka + 1) * HID + col0 + r];
      c = __builtin_amdgcn_wmma_f32_16x16x4_f32(false, a, false, b,
                                                (short)0, c, false, false);
    }

    // C/D layout: VGPR i -> row (i + half*8), col r (within tile)
    float* out = h1 + (size_t)row0 * HID + col0 + r;
#pragma unroll
    for (int i = 0; i < 8; ++i)
      out[(size_t)(i + half * 8) * HID] = c[i];

    wait_async0();       // next tile's LDS writes complete
    __syncthreads();     // all waves done reading cur before it is reused
    cur ^= 1;
  }
}

// ---------------------------------------------------------------- scatter layer 1 (64 feats, 16 lanes/edge x float4)
__global__ void gcn_scatter1(const long long* __restrict__ ei,
                             const float* __restrict__ ea,
                             const float* __restrict__ dinv,
                             const float* __restrict__ h1,
                             float* __restrict__ agg) {
  long long idx = (long long)blockIdx.x * blockDim.x + threadIdx.x;
  const long long total = (long long)(EE + NN) * 16;
  if (idx >= total) return;
  int e = (int)(idx >> 4);
  int t = (int)(idx & 15);
  int s, d; float w;
  if (e < EE) { s = (int)ei[e]; d = (int)ei[EE + e]; w = ea[e]; }
  else        { s = d = e - EE; w = 1.0f; }
  float nrm = dinv[s] * w * dinv[d];
  const float4 hv = *(const float4*)(h1 + (size_t)s * HID + t * 4);
  float* ap = agg + (size_t)d * HID + t * 4;
  atomicAdd(ap + 0, hv.x * nrm);
  atomicAdd(ap + 1, hv.y * nrm);
  atomicAdd(ap + 2, hv.z * nrm);
  atomicAdd(ap + 3, hv.w * nrm);
}

// ---------------------------------------------------------------- bias + ReLU (in place on agg1)
__global__ void gcn_bias_relu(float* __restrict__ agg1, const float* __restrict__ b1) {
  long long i = (long long)blockIdx.x * blockDim.x + threadIdx.x;
  if (i >= (long long)NN * HID) return;
  int f = (int)(i & (HID - 1));
  float v = agg1[i] + b1[f];
  agg1[i] = v > 0.0f ? v : 0.0f;
}

// ---------------------------------------------------------------- GEMM2: h2 = relu(agg1) @ W2  (cols padded 40->48)
// 96 threads = 3 waves; wave w covers cols [16w, 16w+16).
__global__ __launch_bounds__(96) void gcn_gemm2(const float* __restrict__ h,
                                                const float* __restrict__ W2,
                                                float* __restrict__ h2) {
  __shared__ float lw[HID * NC_PAD]; // 12 KB, zero-padded columns
  __shared__ float lx[16 * HID];     // 4 KB
  const int tid  = threadIdx.x;
  const int wave = tid >> 5;         // 0..2
  const int lane = tid & 31;
  const int row0 = blockIdx.x * 16;

  for (int i = tid; i < HID * NC_PAD; i += 96) {
    int k = i / NC_PAD, cc = i % NC_PAD;
    lw[i] = (cc < NCLS) ? W2[k * NCLS + cc] : 0.0f;
  }
  for (int i = tid; i < 16 * HID; i += 96)
    lx[i] = h[(size_t)row0 * HID + i];
  __syncthreads();

  const int half = lane >> 4;
  const int r    = lane & 15;
  const int col0 = wave * 16;

  v8f c = {};
  for (int kb = 0; kb < HID; kb += 4) {
    const int ka = kb + half * 2;
    v2f a, b;
    a.x = lx[r * HID + ka];
    a.y = lx[r * HID + ka + 1];
    b.x = lw[ka * NC_PAD + col0 + r];
    b.y = lw[(ka + 1) * NC_PAD + col0 + r];
    c = __builtin_amdgcn_wmma_f32_16x16x4_f32(false, a, false, b,
                                              (short)0, c, false, false);
  }

  const int col = col0 + r;          // guard AFTER all WMMAs (EXEC all-ones during matrix ops)
  if (col < NCLS) {
    float* out = h2 + (size_t)row0 * NCLS + col;
#pragma unroll
    for (int i = 0; i < 8; ++i)
      out[(size_t)(i + half * 8) * NCLS] = c[i];
  }
}

// ---------------------------------------------------------------- scatter layer 2 (40 feats, 10 lanes/edge x float4)
__global__ void gcn_scatter2(const long long* __restrict__ ei,
                             const float* __restrict__ ea,
                             const float* __restrict__ dinv,
                             const float* __restrict__ h2,
                             float* __restrict__ agg) {
  long long idx = (long long)blockIdx.x * blockDim.x + threadIdx.x;
  const long long total = (long long)(EE + NN) * 10;
  if (idx >= total) return;
  int e = (int)(idx / 10);
  int t = (int)(idx % 10);
  int s, d; float w;
  if (e < EE) { s = (int)ei[e]; d = (int)ei[EE + e]; w = ea[e]; }
  else        { s = d = e - EE; w = 1.0f; }
  float nrm = dinv[s] * w * dinv[d];
  const float4 hv = *(const float4*)(h2 + (size_t)s * NCLS + t * 4);
  float* ap = agg + (size_t)d * NCLS + t * 4;
  atomicAdd(ap + 0, hv.x * nrm);
  atomicAdd(ap + 1, hv.y * nrm);
  atomicAdd(ap + 2, hv.z * nrm);
  atomicAdd(ap + 3, hv.w * nrm);
}

// ---------------------------------------------------------------- bias + log_softmax
__global__ void gcn_logsoftmax(const float* __restrict__ agg2,
                               const float* __restrict__ b2,
                               float* __restrict__ out) {
  int n = blockIdx.x * blockDim.x + threadIdx.x;
  if (n >= NN) return;
  float v[NCLS];
  float m = -INFINITY;
#pragma unroll
  for (int j = 0; j < NCLS; ++j) {
    v[j] = agg2[(size_t)n * NCLS + j] + b2[j];
    m = fmaxf(m, v[j]);
  }
  float s = 0.0f;
#pragma unroll
  for (int j = 0; j < NCLS; ++j) s += __expf(v[j] - m);
  float lse = m + __logf(s);
#pragma unroll
  for (int j = 0; j < NCLS; ++j)
    out[(size_t)n * NCLS + j] = v[j] - lse;
}

// ---------------------------------------------------------------- launch
extern "C" void kernel_launch(void* const* d_in, const int* in_sizes, int n_in,
                              void* d_out, int out_size, void* d_ws, size_t ws_size,
                              hipStream_t stream) {
  (void)in_sizes; (void)n_in; (void)out_size; (void)ws_size;
  const float*     x  = (const float*)d_in[0];
  const long long* ei = (const long long*)d_in[1];   // int64 [2,E]
  const float*     ea = (const float*)d_in[2];
  const float*     W1 = (const float*)d_in[3];
  const float*     b1 = (const float*)d_in[4];
  const float*     W2 = (const float*)d_in[5];
  const float*     b2 = (const float*)d_in[6];
  float* out = (float*)d_out;

  // workspace layout (floats): deg | h1 | agg1 | h2 | agg2
  float* ws   = (float*)d_ws;
  float* deg  = ws;                             // NN           (becomes dinv)
  float* h1   = deg  + NN;                      // NN*HID
  float* agg1 = h1   + (size_t)NN * HID;        // NN*HID
  float* h2   = agg1 + (size_t)NN * HID;        // NN*NCLS
  float* agg2 = h2   + (size_t)NN * NCLS;       // NN*NCLS

  const long long zcount = (long long)NN * (1 + 2 * HID + 2 * NCLS);
  gcn_zero<<<4096, 256, 0, stream>>>(ws, zcount);

  gcn_deg<<<(EE + NN + 255) / 256, 256, 0, stream>>>(ei, ea, deg);
  gcn_dinv<<<(NN + 255) / 256, 256, 0, stream>>>(deg);

  gcn_gemm1<<<NN / (16 * TPB1), 128, 0, stream>>>(x, W1, h1);

  {
    const long long total = (long long)(EE + NN) * 16;
    gcn_scatter1<<<(int)((total + 255) / 256), 256, 0, stream>>>(ei, ea, deg, h1, agg1);
  }
  gcn_bias_relu<<<(int)(((long long)NN * HID + 255) / 256), 256, 0, stream>>>(agg1, b1);

  gcn_gemm2<<<NN / 16, 96, 0, stream>>>(agg1, W2, h2);

  {
    const long long total = (long long)(EE + NN) * 10;
    gcn_scatter2<<<(int)((total + 255) / 256), 256, 0, stream>>>(ei, ea, deg, h2, agg2);
  }
  gcn_logsoftmax<<<(NN + 255) / 256, 256, 0, stream>>>(agg2, b2, out);
}